// fusion_loss_16612933501271
// MI455X (gfx1250) — compile-verified
//
#include <hip/hip_runtime.h>

typedef __attribute__((ext_vector_type(2))) float v2f;
typedef __attribute__((ext_vector_type(8))) float v8f;

#define IMG_H 512
#define IMG_W 512
#define NBATCH 32
#define NPIX (NBATCH * IMG_H * IMG_W)   // 8,388,608
#define NBLOCKS 512
#define TPB 256
#define STRIP 8                         // rows per strip walked by one thread
#define NSTRIPS 8                       // strips per thread: 32*64*512 tasks / 131072 threads

// Load one image row triple at row index `row`, applying scalar row mask rm and
// hoisted x-edge masks. finite*0.0f == 0.0f -> exact zero padding.
__device__ __forceinline__ void ldrow(const float* __restrict__ p, int row, float rm,
                                      int xm, int xc, int xp, float ml, float mr,
                                      float* __restrict__ o) {
    const float* __restrict__ r = p + row * IMG_W;
    o[0] = r[xm] * (rm * ml);
    o[1] = r[xc] *  rm;
    o[2] = r[xp] * (rm * mr);
}

__device__ __forceinline__ float sobel3r(const float* __restrict__ t,
                                         const float* __restrict__ c,
                                         const float* __restrict__ b) {
    const float gx = (t[2] - t[0]) + 2.0f * (c[2] - c[0]) + (b[2] - b[0]);
    const float gy = (t[0] - b[0]) + 2.0f * (t[1] - b[1]) + (t[2] - b[2]);
    return fabsf(gx) + fabsf(gy);
}

__global__ void fusion_loss_main(const float* __restrict__ A,
                                 const float* __restrict__ B,
                                 const float* __restrict__ F,
                                 const int* __restrict__ scheme,
                                 float* __restrict__ partial) {
    __shared__ float wavesum[TPB / 32];

    // task t = tid + si*131072 ; x = t&511 ; q = t>>9 = (blockIdx>>1) + 256*si
    // q is block-uniform -> b, ys, scheme[b] are scalars.
    const int x     = ((blockIdx.x & 1) << 8) | threadIdx.x;   // 0..511 (per-lane)
    const int qbase = blockIdx.x >> 1;                         // 0..255 (scalar)

    // x-side window state: loop-invariant, hoisted
    const int   xm = (x > 0)         ? x - 1 : 0;
    const int   xp = (x < IMG_W - 1) ? x + 1 : IMG_W - 1;
    const float ml = (x > 0)         ? 1.0f : 0.0f;
    const float mr = (x < IMG_W - 1) ? 1.0f : 0.0f;

    float acc = 0.0f;

    for (int si = 0; si < NSTRIPS; ++si) {
        const int q  = qbase + (si << 8);          // scalar
        const int b  = q >> 6;                     // batch image (scalar)
        const int ys = (q & 63) << 3;              // strip top row (scalar)

        const int base = b * (IMG_H * IMG_W);
        const float* __restrict__ pa = A + base;
        const float* __restrict__ pb = B + base;
        const float* __restrict__ pf = F + base;

        const int   s  = scheme[b];                // scalar load
        const float m0 = (s == 0) ? 1.0f : 0.0f;
        const float m1 = (s == 1) ? 1.0f : 0.0f;

        // 4-slot circular row buffers per image. Under unroll-by-4, j mod 4 is a
        // compile-time constant in each unrolled instance, so all slot indices
        // fold, SROA keeps the arrays in registers, and the scheduler's load
        // hoisting window is bounded to ~4 steps (no spills).
        float wa[4][3], wb[4][3], wf[4][3];

        // prime: slot0 = row ys-1 (masked at image top), slot1 = row ys
        const int   yt = (ys > 0) ? ys - 1 : 0;
        const float mt = (ys > 0) ? 1.0f : 0.0f;   // scalar
        ldrow(pa, yt, mt,   xm, x, xp, ml, mr, wa[0]);
        ldrow(pb, yt, mt,   xm, x, xp, ml, mr, wb[0]);
        ldrow(pf, yt, mt,   xm, x, xp, ml, mr, wf[0]);
        ldrow(pa, ys, 1.0f, xm, x, xp, ml, mr, wa[1]);
        ldrow(pb, ys, 1.0f, xm, x, xp, ml, mr, wb[1]);
        ldrow(pf, ys, 1.0f, xm, x, xp, ml, mr, wf[1]);

        #pragma unroll 4
        for (int j = 0; j < STRIP; ++j) {
            const int st = j & 3;                  // top slot    (row ys+j-1)
            const int sc = (j + 1) & 3;            // center slot (row ys+j)
            const int sb = (j + 2) & 3;            // bottom slot (row ys+j+1, loaded now)

            const int y = ys + j;
            // bottom row y+1: in-range for j<7 (mask folds to literal 1.0f);
            // at j==7 it is the next strip's row unless at the image bottom.
            const int   yb  = (j < STRIP - 1) ? y + 1
                              : ((ys < IMG_H - STRIP) ? y + 1 : IMG_H - 1);
            const float mbm = (j < STRIP - 1) ? 1.0f
                              : ((ys < IMG_H - STRIP) ? 1.0f : 0.0f);   // scalar
            ldrow(pa, yb, mbm, xm, x, xp, ml, mr, wa[sb]);
            ldrow(pb, yb, mbm, xm, x, xp, ml, mr, wb[sb]);
            ldrow(pf, yb, mbm, xm, x, xp, ml, mr, wf[sb]);

            const float sA = sobel3r(wa[st], wa[sc], wa[sb]);
            const float sB = sobel3r(wb[st], wb[sc], wb[sb]);
            const float sF = sobel3r(wf[st], wf[sc], wf[sb]);
            const float av = wa[sc][1], bv = wb[sc][1], fv = wf[sc][1]; // centers: valid rows

            const float abv = m0 * (0.5f * (av + bv)) + m1 * fmaxf(av, bv);
            acc += fabsf(abv - fv) + fabsf(sF - fmaxf(sA, sB));
        }
    }

    // ---- exact f32 wave32 reduction via V_WMMA_F32_16X16X4_F32 ----
    // A (16x4 f32): lane L contributes A[L%16][L<16 ? 0 : 2] = acc (other K = 0).
    // B = all-ones (4x16)  =>  D[m][n] = p_m + p_(m+16)  (exact: two nonzero FMAs).
    // EXEC all-ones here: every loop above has a uniform compile-time trip count.
    v2f a2; a2.x = acc;  a2.y = 0.0f;
    v2f b2; b2.x = 1.0f; b2.y = 1.0f;
    v8f c8 = {};
    v8f d = __builtin_amdgcn_wmma_f32_16x16x4_f32(
        /*neg_a=*/false, a2, /*neg_b=*/false, b2,
        /*c_mod=*/(short)0, c8, /*reuse_a=*/false, /*reuse_b=*/false);

    // lanes 0-15 hold rows 0..7 of D; lanes 16-31 hold rows 8..15 (all columns equal)
    float s8 = d[0] + d[1] + d[2] + d[3] + d[4] + d[5] + d[6] + d[7];
    s8 += __shfl_xor(s8, 16, 32);                  // full-wave sum in every lane

    const int lane = threadIdx.x & 31;
    const int wave = threadIdx.x >> 5;
    if (lane == 0) wavesum[wave] = s8;
    __syncthreads();

    if (threadIdx.x == 0) {
        float t = 0.0f;
        for (int i = 0; i < TPB / 32; ++i) t += wavesum[i];
        partial[blockIdx.x] = t;                   // every block writes -> no ws init needed
    }
}

__global__ void fusion_loss_final(const float* __restrict__ partial,
                                  float* __restrict__ out) {
    if (threadIdx.x == 0) {
        float t = 0.0f;
        for (int i = 0; i < NBLOCKS; ++i) t += partial[i];   // fixed order: deterministic
        out[0] = t / (float)NPIX;
    }
}

extern "C" void kernel_launch(void* const* d_in, const int* in_sizes, int n_in,
                              void* d_out, int out_size, void* d_ws, size_t ws_size,
                              hipStream_t stream) {
    (void)in_sizes; (void)n_in; (void)out_size; (void)ws_size;
    const float* A      = (const float*)d_in[0];
    const float* B      = (const float*)d_in[1];
    const float* F      = (const float*)d_in[2];
    const int*   scheme = (const int*)d_in[3];
    float* partial = (float*)d_ws;                 // 512 floats of scratch
    float* out     = (float*)d_out;

    fusion_loss_main<<<NBLOCKS, TPB, 0, stream>>>(A, B, F, scheme, partial);
    fusion_loss_final<<<1, 32, 0, stream>>>(partial, out);
}